// CTCBeamDecoder_35923106463927
// MI455X (gfx1250) — compile-verified
//
#include <hip/hip_runtime.h>
#include <hip/hip_bf16.h>
#include <stdint.h>

#define BW 8
#define TT 256
#define VV 128
#define CH 16
#define CHUNK_ELEMS (CH * VV)   /* 2048 floats = 8 KB per chunk */

#define LOG0f   (-1.0e30f)
#define ACTIVEf (-1.0e29f)
#define PRUNEf  (-9.0f)

typedef unsigned int u32x4 __attribute__((ext_vector_type(4)));
typedef int          i32x8 __attribute__((ext_vector_type(8)));
typedef int          i32x4 __attribute__((ext_vector_type(4)));

__device__ __forceinline__ float lae(float a, float b) {
  float mx = fmaxf(a, b);
  float mn = fminf(a, b);
  return mx + log1pf(expf(mn - mx));
}

// total order: larger value wins; on exact tie, smaller index wins (jax.lax.top_k stability)
__device__ __forceinline__ bool cand_better(float v0, int i0, float v1, int i1) {
  return (v0 > v1) || (v0 == v1 && i0 < i1);
}

__launch_bounds__(32)
__global__ void ctc_beam_kernel(const float* __restrict__ probs,
                                const int* __restrict__ lengths,
                                const int* __restrict__ blankp,
                                float* __restrict__ out) {
  // Manual LDS layout (byte offsets known -> usable as TDM lds_addr; static LDS starts at 0)
  __shared__ __align__(16) unsigned char smem[33792];
  float* probBuf = (float*)(smem);               // [2][2048] double-buffered prob chunks (16 KB)
  int*   pref    = (int*)(smem + 16384);         // [2][8][256] prefix double buffer (16 KB)
  unsigned char* st = smem + 32768;
  float* s_pb     = (float*)(st + 0);
  float* s_pnb    = (float*)(st + 32);
  float* s_ptot   = (float*)(st + 64);
  float* s_contb  = (float*)(st + 96);
  float* s_contnb = (float*)(st + 128);
  float* s_npb    = (float*)(st + 160);
  float* s_npnb   = (float*)(st + 192);
  int*   s_lens   = (int*)(st + 224);
  int*   s_last   = (int*)(st + 256);
  int*   s_par    = (int*)(st + 288);
  int*   s_haspar = (int*)(st + 320);
  int*   s_act    = (int*)(st + 352);
  int*   s_nlens  = (int*)(st + 384);
  int*   s_nsrc   = (int*)(st + 416);
  int*   s_nlab   = (int*)(st + 448);
  int*   s_nisext = (int*)(st + 480);
  int*   s_misc   = (int*)(st + 512);

  const int b    = blockIdx.x;
  const int lane = threadIdx.x;
  const int blank = blankp[0];
  int len = lengths[b];
  len = (len < 1) ? 1 : ((len > TT) ? TT : len);
  const float* pbase = probs + (size_t)b * TT * VV;

  // ---- Tensor Data Mover: 1-D tile of 2048 f32 elements into LDS chunk buffer ----
  auto tdm_issue = [&](int c) {
    unsigned long long ga = (unsigned long long)(uintptr_t)(pbase + (size_t)c * CHUNK_ELEMS);
    unsigned ldsOff = (unsigned)((c & 1) * (CHUNK_ELEMS * 4));
    u32x4 g0 = { 0x1u,                                   // count=1, user mode
                 ldsOff,                                 // lds_addr (bytes)
                 (unsigned)(ga & 0xffffffffull),         // global_addr[31:0]
                 (unsigned)((ga >> 32) & 0x1ffffffull) | 0x80000000u }; // addr[56:32] | type=2
    i32x8 g1 = { (int)(2u << 16),                        // data_size = 4 bytes
                 (int)((CHUNK_ELEMS & 0xffffu) << 16),   // tensor_dim0[15:0]
                 (int)(((CHUNK_ELEMS >> 16) & 0xffffu) | (1u << 16)), // td0 hi | tensor_dim1=1
                 (int)((CHUNK_ELEMS & 0xffffu) << 16),   // td1 hi (0) | tile_dim0 = 2048
                 1,                                      // tile_dim1 = 1, tile_dim2 = 0
                 (int)CHUNK_ELEMS,                       // tensor_dim0_stride lo32
                 0, 0 };
    i32x4 gz4 = { 0, 0, 0, 0 };
    i32x8 gz8 = { 0, 0, 0, 0, 0, 0, 0, 0 };
    __builtin_amdgcn_tensor_load_to_lds(g0, g1, gz4, gz4, gz8, 0);
  };

  // ---- init beam state ----
  {
    int4 m1; m1.x = m1.y = m1.z = m1.w = -1;
    int4* p0 = (int4*)(pref);                 // buffer 0: 8 rows * 64 int4
    #pragma unroll
    for (int it = 0; it < 16; ++it) p0[it * 32 + lane] = m1;
  }
  if (lane < BW) {
    s_pb[lane]   = (lane == 0) ? 0.0f : LOG0f;
    s_pnb[lane]  = LOG0f;
    s_lens[lane] = 0;
  }
  __syncthreads();

  tdm_issue(0);
  const int nch = (len + CH - 1) / CH;
  for (int c = 0; c < nch; ++c) {
    if (c + 1 < nch) {
      tdm_issue(c + 1);
      __builtin_amdgcn_s_wait_tensorcnt(1);   // oldest (chunk c) complete, prefetch in flight
    } else {
      __builtin_amdgcn_s_wait_tensorcnt(0);
    }
    __syncthreads();
    const float* prchunk = probBuf + (c & 1) * CHUNK_ELEMS;
    const int tEnd = (len < (c + 1) * CH) ? len : (c + 1) * CH;

    for (int t = c * CH; t < tEnd; ++t) {
      const float* pr = prchunk + (t - c * CH) * VV;
      const int cur = t & 1;
      int* prefC = pref + cur * (BW * TT);
      int* prefN = pref + (cur ^ 1) * (BW * TT);

      // Phase 1: p_tot, act, last
      if (lane < BW) {
        float pt = lae(s_pb[lane], s_pnb[lane]);
        s_ptot[lane] = pt;
        s_act[lane]  = (pt > ACTIVEf) ? 1 : 0;
        int L = s_lens[lane];
        s_last[lane] = (L > 0) ? prefC[lane * TT + L - 1] : blank;
      }
      __syncthreads();

      // Phase 2: parent match (whole-wave ballot compares, first j = argmax)
      for (int i = 0; i < BW; ++i) {
        int found = -1;
        if (s_act[i]) {
          int Li = s_lens[i];
          for (int j = 0; j < BW; ++j) {
            if (s_act[j] && (s_lens[j] + 1 == Li)) {
              int Lj = s_lens[j];
              bool ok = true;
              for (int base = 0; base < Lj && ok; base += 32) {
                int tp = base + lane;
                bool e = (tp >= Lj) || (prefC[i * TT + tp] == prefC[j * TT + tp]);
                ok = (__all((int)e) != 0);
              }
              if (ok) { found = j; break; }
            }
          }
        }
        if (lane == 0) { s_haspar[i] = (found >= 0) ? 1 : 0; s_par[i] = (found >= 0) ? found : 0; }
      }
      __syncthreads();

      // Phase 3: continuation scores
      if (lane < BW) {
        int i = lane;
        float parprev = LOG0f;
        if (s_haspar[i]) {
          int p = s_par[i];
          parprev = (s_last[i] == s_last[p]) ? s_pb[p] : s_ptot[p];
        }
        s_contnb[i] = (s_lens[i] > 0) ? (lae(s_pnb[i], parprev) + pr[s_last[i]]) : LOG0f;
        s_contb[i]  = s_ptot[i] + pr[blank];
      }
      __syncthreads();

      // Phase 4: candidate generation + per-lane sorted top-8
      float r_pb[BW], r_ptot[BW];
      int r_last[BW], r_act[BW], r_par[BW], r_hp[BW];
      #pragma unroll
      for (int i = 0; i < BW; ++i) {
        r_pb[i] = s_pb[i]; r_ptot[i] = s_ptot[i]; r_last[i] = s_last[i];
        r_act[i] = s_act[i]; r_par[i] = s_par[i]; r_hp[i] = s_haspar[i];
      }
      float tv[BW]; int ti[BW];
      #pragma unroll
      for (int k = 0; k < BW; ++k) { tv[k] = -__builtin_inff(); ti[k] = 0x7fffffff; }
      auto ins = [&](float v, int idx) {
        if (cand_better(v, idx, tv[BW - 1], ti[BW - 1])) {
          tv[BW - 1] = v; ti[BW - 1] = idx;
          #pragma unroll
          for (int k = BW - 1; k > 0; --k) {
            if (cand_better(tv[k], ti[k], tv[k - 1], ti[k - 1])) {
              float fv = tv[k]; tv[k] = tv[k - 1]; tv[k - 1] = fv;
              int fi = ti[k]; ti[k] = ti[k - 1]; ti[k - 1] = fi;
            }
          }
        }
      };
      if (lane < BW) ins(lae(s_contb[lane], s_contnb[lane]), lane);
      #pragma unroll
      for (int k = 0; k < 32; ++k) {            // candidate e = 32k + lane; i = k>>2 is compile-time
        const int i = k >> 2;
        const int v = ((k & 3) << 5) + lane;
        float prv  = pr[v];
        float prev = (v == r_last[i]) ? r_pb[i] : r_ptot[i];
        float extv = prv + prev;
        bool badc = (v == blank) || (prv <= PRUNEf) || (r_act[i] == 0);
        #pragma unroll
        for (int p = 0; p < BW; ++p)
          badc = badc || (r_hp[p] && (r_par[p] == i) && (r_last[p] == v));
        if (badc) extv = LOG0f;
        ins(lae(LOG0f, extv), BW + i * VV + v);
      }

      // Phase 5: 8 rounds of wave32 butterfly arg-max (stable tie-break)
      int selCand = 0;
      #pragma unroll
      for (int r = 0; r < BW; ++r) {
        float bv = tv[0]; int bi = ti[0];
        #pragma unroll
        for (int off = 16; off >= 1; off >>= 1) {
          float ov = __shfl_xor(bv, off, 32);
          int   oi = __shfl_xor(bi, off, 32);
          if (cand_better(ov, oi, bv, bi)) { bv = ov; bi = oi; }
        }
        if (lane == r) selCand = bi;
        if (ti[0] == bi) {                      // winner pops its head
          #pragma unroll
          for (int k = 0; k < BW - 1; ++k) { tv[k] = tv[k + 1]; ti[k] = ti[k + 1]; }
          tv[BW - 1] = -__builtin_inff(); ti[BW - 1] = 0x7fffffff;
        }
      }

      // Phase 6: unpack the 8 winners (lane k handles new beam k)
      if (lane < BW) {
        int cnd  = selCand;
        bool ise = cnd >= BW;
        int e    = cnd - BW;
        int src  = ise ? (e >> 7) : cnd;
        int v    = e & (VV - 1);
        s_nsrc[lane]   = src;
        s_nlab[lane]   = v;
        s_nisext[lane] = ise ? 1 : 0;
        s_nlens[lane]  = s_lens[src] + (ise ? 1 : 0);
        if (ise) {
          float prv  = pr[v];
          float prev = (v == s_last[src]) ? s_pb[src] : s_ptot[src];
          bool badc = (v == blank) || (prv <= PRUNEf) || (s_act[src] == 0);
          for (int p = 0; p < BW; ++p)
            badc = badc || (s_haspar[p] && (s_par[p] == src) && (s_last[p] == v));
          s_npb[lane]  = LOG0f;
          s_npnb[lane] = badc ? LOG0f : (prv + prev);
        } else {
          s_npb[lane]  = s_contb[src];
          s_npnb[lane] = s_contnb[src];
        }
      }
      __syncthreads();

      // Phase 7: prefix copy (8 rows * 64 int4) into the other buffer
      #pragma unroll
      for (int it = 0; it < 16; ++it) {
        int idx = it * 32 + lane;
        int row = idx >> 6;
        int chk = idx & 63;
        const int4* sp = (const int4*)(prefC + s_nsrc[row] * TT);
        int4*       dp = (int4*)(prefN + row * TT);
        dp[chk] = sp[chk];
      }
      __syncthreads();
      if (lane < BW && s_nisext[lane]) {
        prefN[lane * TT + s_lens[s_nsrc[lane]]] = s_nlab[lane];   // old lens[src]
      }
      __syncthreads();
      if (lane < BW) {
        s_lens[lane] = s_nlens[lane];
        s_pb[lane]   = s_npb[lane];
        s_pnb[lane]  = s_npnb[lane];
      }
      __syncthreads();
    }
  }
  __builtin_amdgcn_s_wait_tensorcnt(0);

  // ---- final: argmax of totals (first max), emit sequence + length ----
  if (lane == 0) {
    float bestv = -__builtin_inff(); int best = 0;
    for (int i = 0; i < BW; ++i) {
      float tot = lae(s_pb[i], s_pnb[i]);
      if (tot > bestv) { bestv = tot; best = i; }
    }
    s_misc[0] = best;
  }
  __syncthreads();
  const int best = s_misc[0];
  const int blen = s_lens[best];
  const int* prefF = pref + (len & 1) * (BW * TT) + best * TT;
  #pragma unroll
  for (int it = 0; it < TT / 32; ++it) {
    int tp = it * 32 + lane;
    out[(size_t)b * TT + tp] = (tp < blen) ? (float)prefF[tp] : 0.0f;
  }
  if (lane == 0) out[(size_t)gridDim.x * TT + b] = (float)blen;
}

extern "C" void kernel_launch(void* const* d_in, const int* in_sizes, int n_in,
                              void* d_out, int out_size, void* d_ws, size_t ws_size,
                              hipStream_t stream) {
  const float* probs   = (const float*)d_in[0];
  const int*   lengths = (const int*)d_in[1];
  // d_in[2] = beam_width (fixed 8), d_in[3] = blank_index
  const int*   blank   = (const int*)d_in[3];
  const int B = in_sizes[1];
  ctc_beam_kernel<<<B, 32, 0, stream>>>(probs, lengths, blank, (float*)d_out);
}